// CBPMFModel_34179349742389
// MI455X (gfx1250) — compile-verified
//
#include <hip/hip_runtime.h>

typedef __attribute__((ext_vector_type(2))) float v2f;
typedef __attribute__((ext_vector_type(8))) float v8f;

#define LATENT 128

// One wave (32 lanes) computes 16 batch rows.
// mean[tile] = diag( A(16x128) * B(128x16) ) via 32x V_WMMA_F32_16X16X4_F32.
// f32 WMMA lane layout (A 16x4 and B^T rows coincide here):
//   lane L supplies row (L & 15), K elements {koff, koff+1}, koff = (L>=16)?2:0
// -> a single global_load_b64 per matrix per K-step, same addressing for A and B.
__global__ __launch_bounds__(256) void cbpmf_mean_std_kernel(
    const int*   __restrict__ user_idx,
    const int*   __restrict__ item_idx,
    const float* __restrict__ U,
    const float* __restrict__ V,
    const float* __restrict__ alpha_p,
    const float* __restrict__ gamma_u,
    const float* __restrict__ gamma_v,
    float*       __restrict__ out_mean,
    float*       __restrict__ out_std,
    int batch)
{
    const int lane = threadIdx.x & 31;
    const int wave = threadIdx.x >> 5;

    const long tile_base = ((long)blockIdx.x * (blockDim.x >> 5) + wave) * 16;
    if (tile_base >= batch) return;               // wave-uniform: EXEC stays all-ones

    const int  r        = lane & 15;              // row within tile
    const long row_real = tile_base + r;          // for stores (unclamped)
    const long row      = (row_real < batch) ? row_real : (long)batch - 1; // clamped loads

    const int ui   = user_idx[row];
    const int vi   = item_idx[row];
    const int koff = (lane >> 4) << 1;            // 0 for lanes 0-15, 2 for lanes 16-31

    const float* up = U + (long)ui * LATENT + koff;
    const float* vp = V + (long)vi * LATENT + koff;

    v8f c0 = {0.f, 0.f, 0.f, 0.f, 0.f, 0.f, 0.f, 0.f};
    v8f c1 = {0.f, 0.f, 0.f, 0.f, 0.f, 0.f, 0.f, 0.f};

    // K = 128, 4 per WMMA -> 32 WMMAs; two accumulators break the RAW chain.
    #pragma unroll
    for (int t = 0; t < 32; t += 2) {
        v2f a0 = *(const v2f*)(up + 4 * t);
        v2f b0 = *(const v2f*)(vp + 4 * t);
        v2f a1 = *(const v2f*)(up + 4 * t + 4);
        v2f b1 = *(const v2f*)(vp + 4 * t + 4);
        c0 = __builtin_amdgcn_wmma_f32_16x16x4_f32(false, a0, false, b0,
                                                   (short)0, c0, false, false);
        c1 = __builtin_amdgcn_wmma_f32_16x16x4_f32(false, a1, false, b1,
                                                   (short)0, c1, false, false);
    }
    c0 += c1;

    // Diagonal of D (16x16 f32): (m,m) sits in lane m / VGPR m for m<8,
    // and lane m+16 / VGPR m-8 for m>=8  => holder lanes are 0-7 and 24-31.
    int sel = -1;
    if (lane < 8)        sel = lane;
    else if (lane >= 24) sel = lane - 24;

    float mean = 0.f;
    #pragma unroll
    for (int q = 0; q < 8; ++q)
        if (sel == q) mean = c0[q];

    if (sel >= 0) {
        const int  m  = (lane < 8) ? lane : (lane - 16);
        const long rm = tile_base + m;
        if (rm < batch) out_mean[rm] = mean;
    }

    // std = rsqrt(alpha * gamma_u[u] * gamma_v[i]); lanes 0-15 handle one row each.
    if (lane < 16 && row_real < batch) {
        const float prec = alpha_p[0] * gamma_u[ui] * gamma_v[vi];
        out_std[row_real] = 1.0f / sqrtf(prec);
    }
}

extern "C" void kernel_launch(void* const* d_in, const int* in_sizes, int n_in,
                              void* d_out, int out_size, void* d_ws, size_t ws_size,
                              hipStream_t stream) {
    const int*   user_idx = (const int*)  d_in[0];
    const int*   item_idx = (const int*)  d_in[1];
    const float* U        = (const float*)d_in[2];
    const float* V        = (const float*)d_in[3];
    const float* alpha_p  = (const float*)d_in[4];
    const float* gamma_u  = (const float*)d_in[5];
    const float* gamma_v  = (const float*)d_in[6];

    const int batch = in_sizes[0];
    float* out_mean = (float*)d_out;
    float* out_std  = out_mean + batch;

    const int block = 256;                 // 8 waves -> 128 rows per block
    const int rows_per_block = (block / 32) * 16;
    const int grid = (batch + rows_per_block - 1) / rows_per_block;

    cbpmf_mean_std_kernel<<<grid, block, 0, stream>>>(
        user_idx, item_idx, U, V, alpha_p, gamma_u, gamma_v,
        out_mean, out_std, batch);
}